// GNNBlock_60000693125427
// MI455X (gfx1250) — compile-verified
//
#include <hip/hip_runtime.h>
#include <hip/hip_bf16.h>

typedef __attribute__((ext_vector_type(2))) float v2f;
typedef __attribute__((ext_vector_type(8))) float v8f;

#define C_IN  256
#define C_OUT 128

// ---------------------------------------------------------------------------
// 1) deg[i] = 1.0f  (self-loop)
// ---------------------------------------------------------------------------
__global__ void init_deg_kernel(float* __restrict__ deg, int N) {
    int i = blockIdx.x * blockDim.x + threadIdx.x;
    if (i < N) deg[i] = 1.0f;
}

// ---------------------------------------------------------------------------
// 2) deg[dst] += 1 for each edge
// ---------------------------------------------------------------------------
__global__ void accum_deg_kernel(const long long* __restrict__ ei,
                                 float* __restrict__ deg, int E) {
    int e = blockIdx.x * blockDim.x + threadIdx.x;
    if (e < E) {
        long long dst = ei[(long long)E + e];
        atomicAdd(&deg[dst], 1.0f);
    }
}

// ---------------------------------------------------------------------------
// 3) dis[i] = rsqrt(deg[i])   (deg >= 1 always due to self-loop)
// ---------------------------------------------------------------------------
__global__ void compute_dis_kernel(const float* __restrict__ deg,
                                   float* __restrict__ dis, int N) {
    int i = blockIdx.x * blockDim.x + threadIdx.x;
    if (i < N) dis[i] = rsqrtf(deg[i]);
}

// ---------------------------------------------------------------------------
// 4) h = x @ W  via V_WMMA_F32_16X16X4_F32 (f32 in, f32 accum)
//    Block = 256 threads = 8 waves. Block stages a 32-row M-tile of x in LDS;
//    wave w owns the 16-wide N-tile n0 = 16*w and computes TWO 16x16 output
//    tiles (rows m0..m0+15 and m0+16..m0+31) per k-step, sharing one B
//    fragment -> halves W traffic per WMMA and gives two independent
//    accumulator chains (hides WMMA D->C latency).
//    N % 32 == 0 here (100000/32 = 3125), no tails; EXEC all-ones for WMMA.
//
//    f32 A-frag (16x4): lane<16: row=lane,   K={k,k+1};
//                       lane>=16: row=lane-16, K={k+2,k+3}.
//    f32 B-frag (4x16): lane<16: col=lane,   K={k,k+1};
//                       lane>=16: col=lane-16, K={k+2,k+3}.
//    f32 C/D (16x16): vgpr v: lanes 0-15 -> M=v, lanes 16-31 -> M=v+8; N=lane%16.
// ---------------------------------------------------------------------------
#define LDS_STRIDE 260   // 256 + 4 pad: conflict-free column reads, 16B-aligned rows

__global__ __launch_bounds__(256) void gemm_wmma_kernel(
        const float* __restrict__ x, const float* __restrict__ W,
        float* __restrict__ h) {
    __shared__ float sA[32][LDS_STRIDE];   // ~33.3 KB

    const int m0  = blockIdx.x * 32;
    const int tid = threadIdx.x;

    // Stage 32 x 256 tile of x, float4-vectorized (rows are 1040B => 16B aligned)
    #pragma unroll
    for (int i = tid * 4; i < 32 * C_IN; i += 256 * 4) {
        int r = i >> 8;          // / 256
        int c = i & 255;         // % 256
        float4 v = *(const float4*)(x + (size_t)(m0 + r) * C_IN + c);
        *(float4*)(&sA[r][c]) = v;
    }
    __syncthreads();

    const int wave = tid >> 5;
    const int lane = tid & 31;
    const int n0   = wave * 16;
    const int frag = lane & 15;        // row (A) / col (B,C,D)
    const int kb   = (lane >> 4) * 2;  // K sub-offset within the 4-wide slab

    v8f acc0 = {};
    v8f acc1 = {};
    #pragma unroll 4
    for (int k = 0; k < C_IN; k += 4) {
        v2f af0, af1, bf;
        af0.x = sA[frag][k + kb];
        af0.y = sA[frag][k + kb + 1];
        af1.x = sA[frag + 16][k + kb];
        af1.y = sA[frag + 16][k + kb + 1];
        bf.x  = W[(size_t)(k + kb)     * C_OUT + n0 + frag];
        bf.y  = W[(size_t)(k + kb + 1) * C_OUT + n0 + frag];
        acc0 = __builtin_amdgcn_wmma_f32_16x16x4_f32(
                   false, af0, false, bf, (short)0, acc0, false, false);
        acc1 = __builtin_amdgcn_wmma_f32_16x16x4_f32(
                   false, af1, false, bf, (short)0, acc1, false, false);
    }

    const int n  = n0 + frag;
    const int mb = (lane >> 4) * 8;
    #pragma unroll
    for (int v = 0; v < 8; ++v) {
        h[(size_t)(m0 + mb + v) * C_OUT + n]      = acc0[v];
        h[(size_t)(m0 + 16 + mb + v) * C_OUT + n] = acc1[v];
    }
}

// ---------------------------------------------------------------------------
// 5) zero d_out (harness poisons it; must be re-zeroed deterministically)
// ---------------------------------------------------------------------------
__global__ void zero_out_kernel(float4* __restrict__ out, int n4) {
    int i = blockIdx.x * blockDim.x + threadIdx.x;
    if (i < n4) out[i] = make_float4(0.f, 0.f, 0.f, 0.f);
}

// ---------------------------------------------------------------------------
// 6) scatter: one wave per edge; lane owns 4 channels.
//    out[dst] += h[src] * dis[src]*dis[dst]   (atomic f32, L2-resident)
// ---------------------------------------------------------------------------
__global__ __launch_bounds__(256) void scatter_kernel(
        const float* __restrict__ h, const float* __restrict__ dis,
        const long long* __restrict__ ei, float* __restrict__ out, int E) {
    int t = blockIdx.x * blockDim.x + threadIdx.x;
    int e = t >> 5;
    if (e >= E) return;
    int lane = t & 31;

    long long src = ei[e];
    long long dst = ei[(long long)E + e];
    float nrm = dis[src] * dis[dst];

    float4 v = ((const float4*)(h + src * C_OUT))[lane];
    float* op = out + dst * C_OUT + lane * 4;
    atomicAdd(op + 0, v.x * nrm);
    atomicAdd(op + 1, v.y * nrm);
    atomicAdd(op + 2, v.z * nrm);
    atomicAdd(op + 3, v.w * nrm);
}

// ---------------------------------------------------------------------------
// 7) epilogue: out = relu(out + h*dis^2 (self-loop) + b)
// ---------------------------------------------------------------------------
__global__ void epilogue_kernel(float* __restrict__ out,
                                const float* __restrict__ h,
                                const float* __restrict__ dis,
                                const float* __restrict__ bias, int N) {
    int t = blockIdx.x * blockDim.x + threadIdx.x;     // one float4 per thread
    int total = N * (C_OUT / 4);
    if (t >= total) return;
    int node = t >> 5;          // / 32 float4s per node
    int c4   = t & 31;

    float s  = dis[node];
    float sn = s * s;

    float4 hv = ((const float4*)(h + (size_t)node * C_OUT))[c4];
    float4 bv = ((const float4*)bias)[c4];
    float4* op = (float4*)(out + (size_t)node * C_OUT) + c4;
    float4 ov = *op;

    ov.x = fmaxf(ov.x + hv.x * sn + bv.x, 0.0f);
    ov.y = fmaxf(ov.y + hv.y * sn + bv.y, 0.0f);
    ov.z = fmaxf(ov.z + hv.z * sn + bv.z, 0.0f);
    ov.w = fmaxf(ov.w + hv.w * sn + bv.w, 0.0f);
    *op = ov;
}

// ---------------------------------------------------------------------------
extern "C" void kernel_launch(void* const* d_in, const int* in_sizes, int n_in,
                              void* d_out, int out_size, void* d_ws, size_t ws_size,
                              hipStream_t stream) {
    const float*     x    = (const float*)d_in[0];
    const long long* ei   = (const long long*)d_in[1];   // int64 edge_index [2,E]
    const float*     W    = (const float*)d_in[2];
    const float*     bias = (const float*)d_in[3];
    float*           out  = (float*)d_out;

    const int N = in_sizes[0] / C_IN;   // 100000
    const int E = in_sizes[1] / 2;      // 1600000

    // workspace layout: h [N*128] | deg [N] | dis [N]
    float* h   = (float*)d_ws;
    float* deg = h + (size_t)N * C_OUT;
    float* dis = deg + N;

    init_deg_kernel   <<<(N + 255) / 256, 256, 0, stream>>>(deg, N);
    accum_deg_kernel  <<<(E + 255) / 256, 256, 0, stream>>>(ei, deg, E);
    compute_dis_kernel<<<(N + 255) / 256, 256, 0, stream>>>(deg, dis, N);

    gemm_wmma_kernel  <<<N / 32, 256, 0, stream>>>(x, W, h);

    const int n4 = N * (C_OUT / 4);
    zero_out_kernel   <<<(n4 + 255) / 256, 256, 0, stream>>>((float4*)out, n4);

    const long long sthreads = (long long)E * 32;
    scatter_kernel    <<<(unsigned)((sthreads + 255) / 256), 256, 0, stream>>>(
                          h, dis, ei, out, E);

    epilogue_kernel   <<<(n4 + 255) / 256, 256, 0, stream>>>(out, h, dis, bias, N);
}